// act_net_20255065768379
// MI455X (gfx1250) — compile-verified
//
#include <hip/hip_runtime.h>
#include <math.h>

#define BB 1024
#define NN 100
#define NP 112
#define EE 128
#define MM 8
#define LSTR 132   // padded LDS row stride (floats) to spread banks
#define NT 7       // row tiles (7*16 = 112 >= 100)

typedef __attribute__((ext_vector_type(2))) float v2f;
typedef __attribute__((ext_vector_type(8))) float v8f;

// explicit global-address-space pointers so weight/IO traffic lowers to
// global_load/global_store (LOADcnt/STOREcnt only) instead of flat_load
// (which also ties up DScnt and the LDS pipe on CDNA5)
typedef const float __attribute__((address_space(1)))* gcf;
typedef float __attribute__((address_space(1)))* gf;
typedef const int __attribute__((address_space(1)))* gci;
typedef const unsigned char __attribute__((address_space(1)))* gcu;

struct Params {
    const float* s;
    const float* rest;
    const float* emb_w;
    const float* emb_b;
    const float* wq[3];
    const float* wk[3];
    const float* wv[3];
    const float* wo[3];
    const float* fw_w[3];
    const float* fw_b[3];
    const float* fb_w[3];
    const float* fb_b[3];
    const float* wq_dec;
    const float* wk_dec;
    const float* wv_dec;
    const float* w_dec;
    const float* q_fin;
    const float* k_fin;
    const int* d;
    const unsigned char* mask;
    float* out_x;
    float* out_pro;
};

// A-fragment (16x4 fp32) from LDS, ISA layout: lanes 0-15 -> K=0,1 ; lanes 16-31 -> K=2,3
__device__ inline v2f ld_a(const float* A, int r0, int k0, int lane) {
    int row = r0 + (lane & 15);
    int kk  = k0 + ((lane >> 4) << 1);
    return *(const v2f*)(A + row * LSTR + kk);
}

// B-fragment (4x16 fp32) from global weights (row-major, stride ldw): mirror of A layout
__device__ inline v2f ld_b(gcf W, int ldw, int k0, int c0, int lane) {
    int col = c0 + (lane & 15);
    int kk  = k0 + ((lane >> 4) << 1);
    v2f b;
    b.x = W[kk * ldw + col];
    b.y = W[(kk + 1) * ldw + col];
    return b;
}

// [112 x 128] = A(LDS) @ W(global 128 x ldw slice)  via V_WMMA_F32_16X16X4_F32
// MODE 0: out = acc ; MODE 1: out = tanh(acc + Res) ; MODE 2: out = relu(acc + bias[col])
template <int MODE>
__device__ void gemm_tile(gcf W, int ldw,
                          const float* A, float* Out,
                          gcf bias,
                          const float* Res, int wave, int lane) {
    const int c0 = wave * 16;
    v8f acc[NT];
#pragma unroll
    for (int r = 0; r < NT; ++r) {
        v8f z = {0.f, 0.f, 0.f, 0.f, 0.f, 0.f, 0.f, 0.f};
        acc[r] = z;
    }
    for (int k0 = 0; k0 < EE; k0 += 4) {
        v2f b = ld_b(W, ldw, k0, c0, lane);
#pragma unroll
        for (int r = 0; r < NT; ++r) {
            v2f a = ld_a(A, r * 16, k0, lane);
            acc[r] = __builtin_amdgcn_wmma_f32_16x16x4_f32(
                false, a, false, b, (short)0, acc[r], false, false);
        }
    }
    const int ln  = lane & 15;
    const int rs  = (lane >> 4) << 3;   // lanes >=16 hold rows M+8
    const int col = c0 + ln;
#pragma unroll
    for (int r = 0; r < NT; ++r) {
#pragma unroll
        for (int v = 0; v < 8; ++v) {
            int row = r * 16 + rs + v;
            float x = acc[r][v];
            if (MODE == 1) x = tanhf(x + Res[row * LSTR + col]);
            if (MODE == 2) x = fmaxf(x + bias[col], 0.f);
            Out[row * LSTR + col] = x;
        }
    }
}

__global__ __launch_bounds__(256) void act_net_kernel(Params P) {
    __shared__ float xs[NP * LSTR];
    __shared__ float bufA[NP * LSTR];
    __shared__ float bufB[NP * LSTR];
    __shared__ float sm_avg[EE], sm_qg[EE], sm_ctx[EE], sm_ctx2[EE], sm_qf[EE];
    __shared__ float sm_dsc[MM * NP];
    __shared__ float sm_lg[NP];
    __shared__ float sm_hm[MM], sm_hs[MM];
    __shared__ float sm_red[2];

    const int b    = blockIdx.x;
    const int tid  = threadIdx.x;
    const int wave = tid >> 5;
    const int lane = tid & 31;
    gcu mk = (gcu)P.mask + b * NN;

    // ---- embedding: x = [s | d] @ emb_w + emb_b ----
    {
        gcf sb = (gcf)P.s + (size_t)b * NN * 2;
        gci db = (gci)P.d + (size_t)b * NN;
        gcf ew = (gcf)P.emb_w;
        gcf eb = (gcf)P.emb_b;
        for (int idx = tid; idx < NN * EE; idx += 256) {
            int n = idx >> 7, e = idx & 127;
            float v = sb[n * 2]     * ew[e]
                    + sb[n * 2 + 1] * ew[EE + e]
                    + (float)db[n]  * ew[2 * EE + e]
                    + eb[e];
            xs[n * LSTR + e] = v;
        }
        for (int idx = tid; idx < (NP - NN) * LSTR; idx += 256)
            xs[NN * LSTR + idx] = 0.f;
    }
    __syncthreads();

    // ---- 3 encoder layers ----
    for (int layer = 0; layer < 3; ++layer) {
        // K -> bufA, V -> bufB (WMMA)
        gemm_tile<0>((gcf)P.wk[layer], EE, xs, bufA, nullptr, nullptr, wave, lane);
        gemm_tile<0>((gcf)P.wv[layer], EE, xs, bufB, nullptr, nullptr, wave, lane);
        __syncthreads();

        // attention with online softmax; q computed on the fly
        gcf Wq = (gcf)P.wq[layer];
        float oo[4][16];
#pragma unroll
        for (int j = 0; j < 4; ++j) {
            int p = tid + j * 256;
            if (p < NN * MM) {
                int h = p & 7, n = p >> 3;
                float q[16];
#pragma unroll
                for (int t = 0; t < 16; ++t) q[t] = 0.f;
                const float* xr = xs + n * LSTR;
                for (int e = 0; e < EE; ++e) {
                    float xv = xr[e];
                    gcf wr = Wq + e * EE + h * 16;
#pragma unroll
                    for (int t = 0; t < 16; ++t) q[t] = fmaf(xv, wr[t], q[t]);
                }
                float mx = -3.0e38f, sm = 0.f;
                float o[16];
#pragma unroll
                for (int t = 0; t < 16; ++t) o[t] = 0.f;
                for (int kk = 0; kk < NN; ++kk) {
                    if (mk[kk]) continue;           // masked key -> -inf -> weight 0
                    const float* kr = bufA + kk * LSTR + h * 16;
                    float sc = 0.f;
#pragma unroll
                    for (int t = 0; t < 16; ++t) sc = fmaf(q[t], kr[t], sc);
                    float nm = fmaxf(mx, sc);
                    float fo = __expf(mx - nm);
                    float ew = __expf(sc - nm);
                    sm = sm * fo + ew;
                    const float* vr = bufB + kk * LSTR + h * 16;
#pragma unroll
                    for (int t = 0; t < 16; ++t) o[t] = o[t] * fo + ew * vr[t];
                    mx = nm;
                }
                float inv = 1.f / sm;
#pragma unroll
                for (int t = 0; t < 16; ++t) oo[j][t] = o[t] * inv;
            }
        }
        __syncthreads();   // everyone done reading K (bufA)
#pragma unroll
        for (int j = 0; j < 4; ++j) {
            int p = tid + j * 256;
            if (p < NN * MM) {
                int h = p & 7, n = p >> 3;
#pragma unroll
                for (int t = 0; t < 16; ++t) bufA[n * LSTR + h * 16 + t] = oo[j][t];
            }
        }
        __syncthreads();

        // x2 = tanh(O @ W + x) -> bufB
        gemm_tile<1>((gcf)P.wo[layer], EE, bufA, bufB, nullptr, xs, wave, lane);
        __syncthreads();

        // FFN: acc = relu(x2 @ fw + fb) @ fb_w, K=512 chunked by 128
        v8f facc[NT];
#pragma unroll
        for (int r = 0; r < NT; ++r) {
            v8f z = {0.f, 0.f, 0.f, 0.f, 0.f, 0.f, 0.f, 0.f};
            facc[r] = z;
        }
        const int c0 = wave * 16;
        for (int kc = 0; kc < 4; ++kc) {
            // hc = relu(x2 @ fw[:, kc*128 : +128] + fb[kc*128:]) -> bufA
            gemm_tile<2>((gcf)P.fw_w[layer] + kc * EE, 4 * EE, bufB, bufA,
                         (gcf)P.fw_b[layer] + kc * EE, nullptr, wave, lane);
            __syncthreads();
            // facc += hc @ fb_w[kc*128 : , :]
            gcf W2 = (gcf)P.fb_w[layer] + (size_t)kc * EE * EE;
            for (int k0 = 0; k0 < EE; k0 += 4) {
                v2f bfr = ld_b(W2, EE, k0, c0, lane);
#pragma unroll
                for (int r = 0; r < NT; ++r) {
                    v2f afr = ld_a(bufA, r * 16, k0, lane);
                    facc[r] = __builtin_amdgcn_wmma_f32_16x16x4_f32(
                        false, afr, false, bfr, (short)0, facc[r], false, false);
                }
            }
            __syncthreads();
        }
        // x = tanh(x2 + acc + fb_b)
        {
            gcf Fbb = (gcf)P.fb_b[layer];
            const int ln = lane & 15, rs = (lane >> 4) << 3;
            const int col = c0 + ln;
#pragma unroll
            for (int r = 0; r < NT; ++r)
#pragma unroll
                for (int v = 0; v < 8; ++v) {
                    int row = r * 16 + rs + v;
                    xs[row * LSTR + col] =
                        tanhf(bufB[row * LSTR + col] + facc[r][v] + Fbb[col]);
                }
        }
        __syncthreads();
    }

    // ---- decoder ----
    if (tid < EE) {
        float sum = 0.f, cnt = 0.f;
        for (int n = 0; n < NN; ++n)
            if (!mk[n]) { sum += xs[n * LSTR + tid]; cnt += 1.f; }
        sm_avg[tid] = sum / cnt;
    }
    __syncthreads();
    if (tid < EE) {  // qg = [avg, start, rest] @ wq_dec (257x128)
        float a = 0.f;
        gcf W = (gcf)P.wq_dec;
        for (int i = 0; i < EE; ++i) a = fmaf(sm_avg[i], W[i * EE + tid], a);
        for (int i = 0; i < EE; ++i) a = fmaf(xs[i], W[(EE + i) * EE + tid], a);
        a = fmaf(((gcf)P.rest)[b], W[2 * EE * EE + tid], a);
        sm_qg[tid] = a;
    }
    __syncthreads();
    gemm_tile<0>((gcf)P.wk_dec, EE, xs, bufA, nullptr, nullptr, wave, lane);  // kd
    gemm_tile<0>((gcf)P.wv_dec, EE, xs, bufB, nullptr, nullptr, wave, lane);  // vd
    __syncthreads();
    for (int p = tid; p < NN * MM; p += 256) {
        int h = p & 7, n = p >> 3;
        const float* kr = bufA + n * LSTR + h * 16;
        const float* qr = sm_qg + h * 16;
        float sc = 0.f;
#pragma unroll
        for (int t = 0; t < 16; ++t) sc = fmaf(qr[t], kr[t], sc);
        sm_dsc[h * NP + n] = mk[n] ? -1.0e30f : sc;
    }
    __syncthreads();
    if (tid < MM) {
        float mx = -3.0e38f;
        for (int n = 0; n < NN; ++n) mx = fmaxf(mx, sm_dsc[tid * NP + n]);
        float s = 0.f;
        for (int n = 0; n < NN; ++n) s += __expf(sm_dsc[tid * NP + n] - mx);
        sm_hm[tid] = mx; sm_hs[tid] = s;
    }
    __syncthreads();
    if (tid < EE) {  // ctx
        int h = tid >> 4;
        float mx = sm_hm[h], s = sm_hs[h], c = 0.f;
        for (int n = 0; n < NN; ++n) {
            float w = __expf(sm_dsc[h * NP + n] - mx);
            c = fmaf(w, bufB[n * LSTR + tid], c);
        }
        sm_ctx[tid] = c / s;
    }
    __syncthreads();
    if (tid < EE) {
        float a = 0.f;
        gcf W = (gcf)P.w_dec;
        for (int i = 0; i < EE; ++i) a = fmaf(sm_ctx[i], W[i * EE + tid], a);
        sm_ctx2[tid] = a;
    }
    __syncthreads();
    if (tid < EE) {
        float a = 0.f;
        gcf W = (gcf)P.q_fin;
        for (int i = 0; i < EE; ++i) a = fmaf(sm_ctx2[i], W[i * EE + tid], a);
        sm_qf[tid] = a;
    }
    __syncthreads();
    gemm_tile<0>((gcf)P.k_fin, EE, xs, bufA, nullptr, nullptr, wave, lane);   // kf
    __syncthreads();
    if (tid < NN) {
        const float* kr = bufA + tid * LSTR;
        float a = 0.f;
        for (int e = 0; e < EE; ++e) a = fmaf(sm_qf[e], kr[e], a);
        sm_lg[tid] = mk[tid] ? -1.0e30f : 10.f * tanhf(a);
    }
    __syncthreads();
    if (tid == 0) {
        float mx = -3.0e38f;
        for (int n = 0; n < NN; ++n) mx = fmaxf(mx, sm_lg[n]);
        float s = 0.f;
        for (int n = 0; n < NN; ++n) s += __expf(sm_lg[n] - mx);
        sm_red[0] = mx; sm_red[1] = s;
    }
    __syncthreads();
    if (tid < NN) {
        float pro = mk[tid] ? 0.f : __expf(sm_lg[tid] - sm_red[0]) / sm_red[1];
        ((gf)P.out_pro)[(size_t)b * NN + tid] = pro;
    }
    {
        gf ox = (gf)P.out_x + (size_t)b * NN * EE;
        for (int idx = tid; idx < NN * EE; idx += 256)
            ox[idx] = xs[(idx >> 7) * LSTR + (idx & 127)];
    }
}

extern "C" void kernel_launch(void* const* d_in, const int* in_sizes, int n_in,
                              void* d_out, int out_size, void* d_ws, size_t ws_size,
                              hipStream_t stream) {
    (void)in_sizes; (void)n_in; (void)out_size; (void)d_ws; (void)ws_size;
    Params P;
    int k = 0;
    P.s     = (const float*)d_in[k++];   // 0  s (B,N,2)
    P.rest  = (const float*)d_in[k++];   // 1  rest (B,1)
    P.emb_w = (const float*)d_in[k++];   // 2
    P.emb_b = (const float*)d_in[k++];   // 3
    for (int i = 0; i < 3; ++i) {
        P.wq[i]   = (const float*)d_in[k++];
        P.wk[i]   = (const float*)d_in[k++];
        P.wv[i]   = (const float*)d_in[k++];
        P.wo[i]   = (const float*)d_in[k++];
        P.fw_w[i] = (const float*)d_in[k++];
        P.fw_b[i] = (const float*)d_in[k++];
        P.fb_w[i] = (const float*)d_in[k++];
        P.fb_b[i] = (const float*)d_in[k++];
    }
    P.wq_dec = (const float*)d_in[k++];
    P.wk_dec = (const float*)d_in[k++];
    P.wv_dec = (const float*)d_in[k++];
    P.w_dec  = (const float*)d_in[k++];
    P.q_fin  = (const float*)d_in[k++];
    P.k_fin  = (const float*)d_in[k++];
    P.d      = (const int*)d_in[k++];            // 34 d (B,N,1) int32
    P.mask   = (const unsigned char*)d_in[k++];  // 35 mask (B,N) bool
    P.out_x   = (float*)d_out;
    P.out_pro = (float*)d_out + (size_t)BB * NN * EE;

    act_net_kernel<<<dim3(BB), dim3(256), 0, stream>>>(P);
}